// GATEncoder_60120952209955
// MI455X (gfx1250) — compile-verified
//
#include <hip/hip_runtime.h>
#include <math.h>

// ---------------- problem constants (from reference) ----------------
#define HEADS     8
#define HIDC      32
#define L1C       256        // HEADS*HIDC
#define OUTC      16
#define NEG_SLOPE 0.2f

typedef __attribute__((ext_vector_type(2))) float v2f;
typedef __attribute__((ext_vector_type(8))) float v8f;

// monotonic float<->uint encoding so atomicMax(u32) == float max.
// init value 0 acts as -infinity (all finite encodings are > 0, and every
// dst node has a self-loop so the max is always written).
__device__ __forceinline__ unsigned fenc(float f) {
    unsigned u = __float_as_uint(f);
    return (u & 0x80000000u) ? ~u : (u | 0x80000000u);
}
__device__ __forceinline__ float fdec(unsigned k) {
    return __uint_as_float((k & 0x80000000u) ? (k ^ 0x80000000u) : ~k);
}
__device__ __forceinline__ float lrelu(float v) {
    return v > 0.0f ? v : NEG_SLOPE * v;
}
__device__ __forceinline__ void edge_sd(const int* __restrict__ ei, int E, int e,
                                        int& s, int& d) {
    if (e < E) { s = ei[e]; d = ei[E + e]; }
    else       { s = e - E; d = e - E; }          // appended self-loops
}

// ---------------- init ----------------
__global__ void k_fill_zero(float* __restrict__ p, size_t n) {
    size_t i = (size_t)blockIdx.x * blockDim.x + threadIdx.x;
    size_t stride = (size_t)gridDim.x * blockDim.x;
    for (; i < n; i += stride) p[i] = 0.0f;
}

// ---------------- layer 1 GEMM: h1 = x(N,3) @ W1(3,256), WMMA f32 16x16x4 (K padded to 4)
// one wave per 16x16 output tile; 3125 mtiles x 16 ntiles = 50000 waves
__global__ void k_gemm1_wmma(const float* __restrict__ x, const float* __restrict__ W1,
                             float* __restrict__ h1, int n_nodes) {
    int wave   = (int)((blockIdx.x * blockDim.x + threadIdx.x) >> 5);
    int lane   = threadIdx.x & 31;
    int mtiles = (n_nodes + 15) >> 4;
    int mtile  = wave >> 4;           // 16 ntiles per mtile
    int ntile  = wave & 15;
    if (mtile >= mtiles) return;      // wave-uniform
    int m  = lane & 15;
    int kh = lane >> 4;               // K half: lanes 0-15 -> K{0,1}, 16-31 -> K{2,3}
    bool full = (mtile * 16 + 16 <= n_nodes);    // wave-uniform fast path
    int node = mtile * 16 + m;
    if (node >= n_nodes) node = n_nodes - 1;     // clamp (tail only; harmless dup read)
    // Branch-free fragment setup: load all of K=3, select by kh (v_cndmask, no exec branches).
    float x0 = x[node * 3 + 0], x1 = x[node * 3 + 1], x2 = x[node * 3 + 2];
    int col = ntile * 16 + m;
    float w0 = W1[0 * L1C + col], w1 = W1[1 * L1C + col], w2 = W1[2 * L1C + col];
    v2f a, b;
    a.x = kh ? x2 : x0;  a.y = kh ? 0.0f : x1;   // A 16x4 (K padded with 0)
    b.x = kh ? w2 : w0;  b.y = kh ? 0.0f : w1;   // B 4x16
    v8f c = {};
    c = __builtin_amdgcn_wmma_f32_16x16x4_f32(false, a, false, b, (short)0, c,
                                              false, false);
    // C/D layout: lane = N (mod 16), VGPR v = row (v + 8*kh). Single base + imm offsets.
    float* p = h1 + (size_t)(mtile * 16 + 8 * kh) * L1C + col;
    if (full) {
#pragma unroll
        for (int v = 0; v < 8; ++v) p[(size_t)v * L1C] = c[v];
    } else {
        for (int v = 0; v < 8; ++v)
            if (mtile * 16 + 8 * kh + v < n_nodes) p[(size_t)v * L1C] = c[v];
    }
}

// ---------------- per-node attention coefficients, layer 1 ----------------
// block = 256 threads = 8 waves; block per node, wave = head, lane = channel
__global__ void k_att1(const float* __restrict__ h1, const float* __restrict__ as1,
                       const float* __restrict__ ad1, float* __restrict__ a_s,
                       float* __restrict__ a_d, int n_nodes) {
    int node = blockIdx.x;
    if (node >= n_nodes) return;
    int head = threadIdx.x >> 5;
    int lane = threadIdx.x & 31;
    float v = h1[(size_t)node * L1C + head * HIDC + lane];
    float s = v * as1[head * HIDC + lane];
    float d = v * ad1[head * HIDC + lane];
    for (int off = 16; off; off >>= 1) {
        s += __shfl_xor(s, off, 32);
        d += __shfl_xor(d, off, 32);
    }
    if (lane == 0) {
        a_s[node * HEADS + head] = s;
        a_d[node * HEADS + head] = d;
    }
}

// ---------------- edge softmax pass 1: per-(dst,head) max ----------------
__global__ void k_edge_max1(const int* __restrict__ ei, int E, int ET,
                            const float* __restrict__ a_s, const float* __restrict__ a_d,
                            unsigned* __restrict__ m1) {
    int e = blockIdx.x * blockDim.x + threadIdx.x;
    if (e >= ET) return;
    int s, d; edge_sd(ei, E, e, s, d);
#pragma unroll
    for (int h = 0; h < HEADS; ++h) {
        float v = lrelu(a_s[s * HEADS + h] + a_d[d * HEADS + h]);
        atomicMax(&m1[d * HEADS + h], fenc(v));
    }
}

// ---------------- edge softmax pass 2: denominators ----------------
__global__ void k_edge_exp1(const int* __restrict__ ei, int E, int ET,
                            const float* __restrict__ a_s, const float* __restrict__ a_d,
                            const unsigned* __restrict__ m1, float* __restrict__ denom1) {
    int e = blockIdx.x * blockDim.x + threadIdx.x;
    if (e >= ET) return;
    int s, d; edge_sd(ei, E, e, s, d);
#pragma unroll
    for (int h = 0; h < HEADS; ++h) {
        float v = lrelu(a_s[s * HEADS + h] + a_d[d * HEADS + h]);
        float w = expf(v - fdec(m1[d * HEADS + h]));
        atomicAdd(&denom1[d * HEADS + h], w);
    }
}

// ---------------- edge message pass, layer 1 ----------------
// one wave per edge; lane = channel within head; 8 heads looped.
// h1 (51MB) and out1 (51MB) both live in the 192MB L2 -> atomics resolve at L2.
__global__ void k_msg1(const int* __restrict__ ei, int E, int ET,
                       const float* __restrict__ a_s, const float* __restrict__ a_d,
                       const unsigned* __restrict__ m1, const float* __restrict__ denom1,
                       const float* __restrict__ h1, float* __restrict__ out1) {
    long long gid = (long long)blockIdx.x * blockDim.x + threadIdx.x;
    int e = (int)(gid >> 5);
    int lane = (int)(gid & 31);
    if (e >= ET) return;
    int s, d; edge_sd(ei, E, e, s, d);
    float myalpha = 0.0f;
    if (lane < HEADS) {
        float v = lrelu(a_s[s * HEADS + lane] + a_d[d * HEADS + lane]);
        float w = expf(v - fdec(m1[d * HEADS + lane]));
        myalpha = w / (denom1[d * HEADS + lane] + 1e-16f);
    }
    const float* ph = h1 + (size_t)s * L1C + lane;
    float* po = out1 + (size_t)d * L1C + lane;
#pragma unroll
    for (int h = 0; h < HEADS; ++h) {
        float ah = __shfl(myalpha, h, 32);
        atomicAdd(po + h * HIDC, ph[h * HIDC] * ah);
    }
}

// ---------------- bias + ELU (in place) ----------------
__global__ void k_elu(float* __restrict__ out1, const float* __restrict__ bias1, size_t n) {
    size_t i = (size_t)blockIdx.x * blockDim.x + threadIdx.x;
    size_t stride = (size_t)gridDim.x * blockDim.x;
    for (; i < n; i += stride) {
        float v = out1[i] + bias1[i & (L1C - 1)];
        out1[i] = v > 0.0f ? v : (expf(v) - 1.0f);
    }
}

// ---------------- layer 2 GEMM: h2 = helu(N,256) @ W2(256,16), WMMA f32 16x16x4
// one wave per 16-node tile; K=256 -> 64 chained WMMAs
__global__ void k_gemm2_wmma(const float* __restrict__ helu, const float* __restrict__ W2,
                             float* __restrict__ h2, int n_nodes) {
    int wave = (int)((blockIdx.x * blockDim.x + threadIdx.x) >> 5);
    int lane = threadIdx.x & 31;
    int mtiles = (n_nodes + 15) >> 4;
    if (wave >= mtiles) return;       // wave-uniform
    int m  = lane & 15;
    int kh = lane >> 4;
    bool full = (wave * 16 + 16 <= n_nodes);
    int node = wave * 16 + m;
    if (node >= n_nodes) node = n_nodes - 1;
    const float* arow = helu + (size_t)node * L1C + kh * 2;   // this lane's K pair base
    const float* brow = W2 + kh * 2 * OUTC + m;
    v8f c = {};
#pragma unroll 8
    for (int kk = 0; kk < L1C; kk += 4) {
        v2f a = *(const v2f*)(arow + kk);          // 8B-aligned -> global_load_b64
        v2f b;
        b.x = brow[kk * OUTC];
        b.y = brow[kk * OUTC + OUTC];
        c = __builtin_amdgcn_wmma_f32_16x16x4_f32(false, a, false, b, (short)0, c,
                                                  false, false);
    }
    float* p = h2 + (size_t)(wave * 16 + 8 * kh) * OUTC + m;
    if (full) {
#pragma unroll
        for (int v = 0; v < 8; ++v) p[(size_t)v * OUTC] = c[v];
    } else {
        for (int v = 0; v < 8; ++v)
            if (wave * 16 + 8 * kh + v < n_nodes) p[(size_t)v * OUTC] = c[v];
    }
}

// ---------------- per-node attention coefficients, layer 2 (1 head, 16 ch) ----------------
__global__ void k_att2(const float* __restrict__ h2, const float* __restrict__ as2,
                       const float* __restrict__ ad2, float* __restrict__ a_s,
                       float* __restrict__ a_d, int n_nodes) {
    int n = blockIdx.x * blockDim.x + threadIdx.x;
    if (n >= n_nodes) return;
    const float* row = h2 + (size_t)n * OUTC;
    float s = 0.0f, d = 0.0f;
#pragma unroll
    for (int cc = 0; cc < OUTC; ++cc) {
        float v = row[cc];
        s += v * as2[cc];
        d += v * ad2[cc];
    }
    a_s[n] = s;
    a_d[n] = d;
}

__global__ void k_edge_max2(const int* __restrict__ ei, int E, int ET,
                            const float* __restrict__ a_s, const float* __restrict__ a_d,
                            unsigned* __restrict__ m2) {
    int e = blockIdx.x * blockDim.x + threadIdx.x;
    if (e >= ET) return;
    int s, d; edge_sd(ei, E, e, s, d);
    float v = lrelu(a_s[s] + a_d[d]);
    atomicMax(&m2[d], fenc(v));
}

__global__ void k_edge_exp2(const int* __restrict__ ei, int E, int ET,
                            const float* __restrict__ a_s, const float* __restrict__ a_d,
                            const unsigned* __restrict__ m2, float* __restrict__ denom2) {
    int e = blockIdx.x * blockDim.x + threadIdx.x;
    if (e >= ET) return;
    int s, d; edge_sd(ei, E, e, s, d);
    float v = lrelu(a_s[s] + a_d[d]);
    atomicAdd(&denom2[d], expf(v - fdec(m2[d])));
}

// 16 lanes per edge, lane = output channel
__global__ void k_msg2(const int* __restrict__ ei, int E, int ET,
                       const float* __restrict__ a_s, const float* __restrict__ a_d,
                       const unsigned* __restrict__ m2, const float* __restrict__ denom2,
                       const float* __restrict__ h2, float* __restrict__ out) {
    long long gid = (long long)blockIdx.x * blockDim.x + threadIdx.x;
    int e = (int)(gid >> 4);
    int c = (int)(gid & 15);
    if (e >= ET) return;
    int s, d; edge_sd(ei, E, e, s, d);
    float v = lrelu(a_s[s] + a_d[d]);
    float w = expf(v - fdec(m2[d]));
    float alpha = w / (denom2[d] + 1e-16f);
    atomicAdd(&out[(size_t)d * OUTC + c], h2[(size_t)s * OUTC + c] * alpha);
}

__global__ void k_bias2(float* __restrict__ out, const float* __restrict__ bias2, size_t n) {
    size_t i = (size_t)blockIdx.x * blockDim.x + threadIdx.x;
    size_t stride = (size_t)gridDim.x * blockDim.x;
    for (; i < n; i += stride) out[i] += bias2[i & (OUTC - 1)];
}

// ---------------- launch ----------------
extern "C" void kernel_launch(void* const* d_in, const int* in_sizes, int n_in,
                              void* d_out, int out_size, void* d_ws, size_t ws_size,
                              hipStream_t stream) {
    const float* x     = (const float*)d_in[0];
    const int*   ei    = (const int*)d_in[1];
    const float* W1    = (const float*)d_in[2];
    const float* as1   = (const float*)d_in[3];
    const float* ad1   = (const float*)d_in[4];
    const float* bias1 = (const float*)d_in[5];
    const float* W2    = (const float*)d_in[6];
    const float* as2   = (const float*)d_in[7];
    const float* ad2   = (const float*)d_in[8];
    const float* bias2 = (const float*)d_in[9];
    float* out = (float*)d_out;

    const int N  = in_sizes[0] / 3;       // 50000
    const int E  = in_sizes[1] / 2;       // 800000
    const int ET = E + N;                 // edges + self-loops

    // ---- workspace layout (floats); zero-init region is contiguous at front ----
    float* ws = (float*)d_ws;
    const size_t Nz = (size_t)N;
    float*    out1   = ws;                                   // 256N (zeroed)
    unsigned* m1     = (unsigned*)(ws + 256 * Nz);           // 8N   (zeroed)
    float*    denom1 = ws + 264 * Nz;                        // 8N   (zeroed)
    unsigned* m2     = (unsigned*)(ws + 272 * Nz);           // N    (zeroed)
    float*    denom2 = ws + 273 * Nz;                        // N    (zeroed)
    const size_t zero_count = 274 * Nz;
    float*    h1   = ws + 274 * Nz;                          // 256N
    float*    a_s1 = ws + 530 * Nz;                          // 8N
    float*    a_d1 = ws + 538 * Nz;                          // 8N
    float*    h2   = ws + 546 * Nz;                          // 16N
    float*    a_s2 = ws + 562 * Nz;                          // N
    float*    a_d2 = ws + 563 * Nz;                          // N   (total 564N floats)

    const int TB = 256;
    // init (d_out is poisoned by harness; re-zero every call for graph replay)
    k_fill_zero<<<2048, TB, 0, stream>>>(ws, zero_count);
    k_fill_zero<<<512, TB, 0, stream>>>(out, (size_t)N * OUTC);

    // layer 1
    {
        int mtiles = (N + 15) / 16;
        long long waves = (long long)mtiles * 16;
        int blocks = (int)((waves * 32 + TB - 1) / TB);
        k_gemm1_wmma<<<blocks, TB, 0, stream>>>(x, W1, h1, N);
    }
    k_att1<<<N, TB, 0, stream>>>(h1, as1, ad1, a_s1, a_d1, N);
    k_edge_max1<<<(ET + TB - 1) / TB, TB, 0, stream>>>(ei, E, ET, a_s1, a_d1, m1);
    k_edge_exp1<<<(ET + TB - 1) / TB, TB, 0, stream>>>(ei, E, ET, a_s1, a_d1, m1, denom1);
    {
        long long thr = (long long)ET * 32;
        k_msg1<<<(int)((thr + TB - 1) / TB), TB, 0, stream>>>(ei, E, ET, a_s1, a_d1,
                                                              m1, denom1, h1, out1);
    }
    k_elu<<<2048, TB, 0, stream>>>(out1, bias1, (size_t)N * L1C);

    // layer 2
    {
        int mtiles = (N + 15) / 16;
        long long thr = (long long)mtiles * 32;
        k_gemm2_wmma<<<(int)((thr + TB - 1) / TB), TB, 0, stream>>>(out1, W2, h2, N);
    }
    k_att2<<<(N + TB - 1) / TB, TB, 0, stream>>>(h2, as2, ad2, a_s2, a_d2, N);
    k_edge_max2<<<(ET + TB - 1) / TB, TB, 0, stream>>>(ei, E, ET, a_s2, a_d2, m2);
    k_edge_exp2<<<(ET + TB - 1) / TB, TB, 0, stream>>>(ei, E, ET, a_s2, a_d2, m2, denom2);
    {
        long long thr = (long long)ET * 16;
        k_msg2<<<(int)((thr + TB - 1) / TB), TB, 0, stream>>>(ei, E, ET, a_s2, a_d2,
                                                              m2, denom2, h2, out);
    }
    k_bias2<<<512, TB, 0, stream>>>(out, bias2, (size_t)N * OUTC);
}